// PolymorphicNeuron_17901423690365
// MI455X (gfx1250) — compile-verified
//
#include <hip/hip_runtime.h>
#include <math.h>

typedef __attribute__((ext_vector_type(2))) float v2f;
typedef __attribute__((ext_vector_type(4))) float f4;
typedef __attribute__((ext_vector_type(8))) float v8f;

#define THRESH 1.0f

// ---------------------------------------------------------------------------
// Kernel 1: mode_scores = softmax(x @ W^T + b)  via V_WMMA_F32_16X16X4_F32
//   x: [B, D] row-major, W: [M, D] row-major (so B-matrix = W^T column tile)
//   One block = 16 waves (512 threads). Block computes a 16-row x 256-col slab.
//   Wave w computes the 16x16 tile at cols [16w, 16w+16).
//
//   WMMA f32 16x16x4 VGPR layouts (ISA 7.12.2):
//     A (16x4):  lane L<16 : v0=A[L,0], v1=A[L,1];  lane>=16: v0=A[L-16,2], v1=A[L-16,3]
//     B (4x16):  lane L<16 : v0=B[0,L], v1=B[1,L];  lane>=16: v0=B[2,L-16], v1=B[3,L-16]
//     C/D:       vgpr r, lane L: row = r + 8*(L>>4), col = L&15
// ---------------------------------------------------------------------------
__global__ __launch_bounds__(512) void pn_scores_kernel(
    const float* __restrict__ x, const float* __restrict__ W,
    const float* __restrict__ bias, float* __restrict__ scores,
    int B, int D, int M)
{
    __shared__ float sS[16 * 256];          // 16 rows x M cols slab (M == 256)

    const int lane = threadIdx.x & 31;
    const int wave = threadIdx.x >> 5;      // 0..15
    const int rb   = blockIdx.x * 16;       // base row in B
    const int cw   = wave * 16;             // base col in M
    const int mrow = lane & 15;
    const int half = lane >> 4;             // 0 or 1

    v8f acc = {};
    const float* xrow = x + (size_t)(rb + mrow) * D;
    const float* wrow = W + (size_t)(cw + mrow) * D;
    for (int k0 = 0; k0 < D; k0 += 4) {
        const int kk = k0 + 2 * half;
        v2f a, bf;
        a.x  = xrow[kk];
        a.y  = xrow[kk + 1];
        bf.x = wrow[kk];        // B[kk, n] = W[n, kk]
        bf.y = wrow[kk + 1];
        acc = __builtin_amdgcn_wmma_f32_16x16x4_f32(
            /*neg_a=*/false, a, /*neg_b=*/false, bf,
            /*c_mod=*/(short)0, acc, /*reuse_a=*/false, /*reuse_b=*/false);
    }

    // Add bias, stash tile into LDS slab (bias index = cw + (lane&15) for both halves)
    const float bn = bias[cw + mrow];
#pragma unroll
    for (int r = 0; r < 8; ++r) {
        const int row = r + 8 * half;       // 0..15 local row
        sS[row * 256 + (cw + mrow)] = acc[r] + bn;
    }
    __syncthreads();

    // Row softmax over M=256: wave w owns local row w; 8 values per lane.
    {
        const int row = wave;
        float v[8];
        float mx = -INFINITY;
#pragma unroll
        for (int i = 0; i < 8; ++i) {
            v[i] = sS[row * 256 + lane + 32 * i];
            mx = fmaxf(mx, v[i]);
        }
        for (int off = 16; off > 0; off >>= 1)
            mx = fmaxf(mx, __shfl_xor(mx, off, 32));
        float sum = 0.0f;
#pragma unroll
        for (int i = 0; i < 8; ++i) {
            v[i] = expf(v[i] - mx);
            sum += v[i];
        }
        for (int off = 16; off > 0; off >>= 1)
            sum += __shfl_xor(sum, off, 32);
        const float inv = 1.0f / sum;
        float* out = scores + (size_t)(rb + row) * M;
#pragma unroll
        for (int i = 0; i < 8; ++i)
            out[lane + 32 * i] = v[i] * inv;
    }
}

// ---------------------------------------------------------------------------
// Kernel 2: streaming pass (bandwidth bound: 268 MB in + 268 MB out).
//   mem_new[m,b,d] = m*mem[m,b,d] + x[b,d] - (mem[m,b,d] > 1)
//   mixed[b,d]     = sum_m (mem_new[m,b,d] > 1) * scores[b,m]
//   One thread owns a float4 chunk of one (b,:) row; block = 256 threads
//   = 4 consecutive b rows. Scores for those 4 rows staged in LDS.
//   Non-temporal b128 loads/stores on the single-touch 268MB streams.
// ---------------------------------------------------------------------------
__global__ __launch_bounds__(256) void pn_leaky_kernel(
    const float* __restrict__ x, const float* __restrict__ mem_states,
    const float* __restrict__ scores, float* __restrict__ mixed,
    float* __restrict__ mem_new, int B, int D, int M)
{
    __shared__ float sSc[4 * 256];          // scores for this block's 4 rows

    const int tid      = threadIdx.x;
    const int rowInBlk = tid >> 6;          // 0..3  (D/4 = 64 threads per row)
    const int c4       = tid & 63;          // float4 column index
    const int b        = blockIdx.x * 4 + rowInBlk;

    // Stage 4 contiguous score rows (4KB) into LDS.
    for (int i = tid; i < 4 * M; i += 256)
        sSc[i] = scores[(size_t)(blockIdx.x * 4) * M + i];
    __syncthreads();

    const size_t bd4 = (size_t)B * (D / 4); // slab size in float4 units
    const size_t idx = (size_t)b * (D / 4) + c4;

    const f4* xv  = (const f4*)x;
    const f4* msv = (const f4*)mem_states;
    f4*       mnv = (f4*)mem_new;
    f4*       mxv = (f4*)mixed;

    const f4 xr = xv[idx];
    f4 acc = {0.0f, 0.0f, 0.0f, 0.0f};

    for (int m = 0; m < M; ++m) {
        const f4 ms = __builtin_nontemporal_load(&msv[(size_t)m * bd4 + idx]);
        const float beta = (float)m;
        const float s    = sSc[rowInBlk * 256 + m];

        f4 mn;
        mn.x = fmaf(beta, ms.x, xr.x) - ((ms.x > THRESH) ? THRESH : 0.0f);
        mn.y = fmaf(beta, ms.y, xr.y) - ((ms.y > THRESH) ? THRESH : 0.0f);
        mn.z = fmaf(beta, ms.z, xr.z) - ((ms.z > THRESH) ? THRESH : 0.0f);
        mn.w = fmaf(beta, ms.w, xr.w) - ((ms.w > THRESH) ? THRESH : 0.0f);

        __builtin_nontemporal_store(mn, &mnv[(size_t)m * bd4 + idx]);

        acc.x += (mn.x > THRESH) ? s : 0.0f;
        acc.y += (mn.y > THRESH) ? s : 0.0f;
        acc.z += (mn.z > THRESH) ? s : 0.0f;
        acc.w += (mn.w > THRESH) ? s : 0.0f;
    }
    mxv[idx] = acc;
}

// ---------------------------------------------------------------------------
// d_in:  [0]=x (B*D), [1]=mem_states (M*B*D), [2]=W (M*D), [3]=b (M)
// d_out: [mixed (B*D) | mem_new (M*B*D) | mode_scores (B*M)]  (f32)
// ---------------------------------------------------------------------------
extern "C" void kernel_launch(void* const* d_in, const int* in_sizes, int n_in,
                              void* d_out, int out_size, void* d_ws, size_t ws_size,
                              hipStream_t stream)
{
    const float* x   = (const float*)d_in[0];
    const float* mem = (const float*)d_in[1];
    const float* W   = (const float*)d_in[2];
    const float* bia = (const float*)d_in[3];

    const int M = in_sizes[3];            // 256
    const int D = in_sizes[2] / M;        // 256
    const int B = in_sizes[0] / D;        // 1024

    float* mixed   = (float*)d_out;
    float* mem_new = mixed + (size_t)B * D;
    float* scores  = mem_new + (size_t)M * B * D;

    // Pass 1: WMMA GEMM + softmax -> scores (also the 3rd output)
    pn_scores_kernel<<<B / 16, 512, 0, stream>>>(x, W, bia, scores, B, D, M);

    // Pass 2: streaming leaky-integrate / spike / mix
    pn_leaky_kernel<<<B / 4, 256, 0, stream>>>(x, mem, scores, mixed, mem_new, B, D, M);
}